// PostProcessHOITriplet_48378511622690
// MI455X (gfx1250) — compile-verified
//
#include <hip/hip_runtime.h>
#include <stdint.h>

// ---------------------------------------------------------------------------
// PostProcessHOITriplet for MI455X (gfx1250, wave32).
// fp32 throughout (threshold comparisons are bit-sensitive).
// CDNA5 paths:
//   * V_WMMA_F32_16X16X4_F32 performs the softmax row-sum reduction
//     (A = 16x4 chunk of exp values, B = ones) on the matrix pipe.
//   * GLOBAL_LOAD_ASYNC_TO_LDS_B128 (+ s_wait_asynccnt) stages the logits
//     tile directly into LDS, tracked by ASYNCcnt (guarded by __has_builtin,
//     falls back to plain loads if the builtin is unavailable).
// ---------------------------------------------------------------------------

namespace {
constexpr int      QN      = 16384;
constexpr int      CN      = 600;
constexpr int      CPAD    = 608;        // 152 K-chunks of 4 for WMMA
constexpr int      ROWS    = 16;         // rows per score block
constexpr int      KTOP    = 4096;
constexpr unsigned CAP     = 1u << 19;   // candidate capacity (worst case <= 20*QN = 327680)
constexpr float    THR     = 0.05f;
constexpr float    IOU_THR = 0.5f;
constexpr int      NFLAGS  = 8192;       // flat-index flags used to rebuild -inf tail of top_k
constexpr int      NCHUNK  = KTOP / 32;  // 128 u32 mask words per NMS row
}

typedef __attribute__((ext_vector_type(2))) float v2f;
typedef __attribute__((ext_vector_type(8))) float v8f;
typedef int b128_t __attribute__((vector_size(4 * sizeof(int))));  // async-copy payload

__device__ __forceinline__ uint32_t fmap(float f) {       // order-preserving float->u32
  uint32_t b = __float_as_uint(f);
  return (b & 0x80000000u) ? ~b : (b | 0x80000000u);
}
__device__ __forceinline__ float funmap(uint32_t u) {
  return __uint_as_float((u & 0x80000000u) ? (u & 0x7fffffffu) : ~u);
}

// ---------------------------------------------------------------- init ----
__global__ void hoi_init(uint64_t* keys, unsigned* count, unsigned* maxPm,
                         unsigned* maxOm, unsigned* above32) {
  unsigned i = blockIdx.x * blockDim.x + threadIdx.x;
  if (i < CAP) keys[i] = 0ull;
  if (i < NFLAGS / 4) above32[i] = 0u;
  if (i == 0) { *count = 0u; *maxPm = 0u; *maxOm = 0u; }
}

// ------------------------------------------------ softmax * sigmoid ------
// One block = 16 query rows. Row sums via V_WMMA_F32_16X16X4_F32.
__global__ __launch_bounds__(256) void hoi_score(
    const float* __restrict__ logits, const float* __restrict__ box_scores,
    uint64_t* __restrict__ keys, unsigned* __restrict__ count,
    unsigned char* __restrict__ above) {
  __shared__ float tile[ROWS][CPAD];
  __shared__ float rowmax[ROWS];
  __shared__ float rowsum[ROWS];
  __shared__ float sigm[ROWS];

  const int tid = threadIdx.x;
  const int q0  = blockIdx.x * ROWS;

  if (tid < ROWS) sigm[tid] = 1.0f / (1.0f + __expf(-box_scores[q0 + tid]));

#if __has_builtin(__builtin_amdgcn_global_load_async_to_lds_b128)
  // CDNA5 async copy: memory -> LDS, 16B per lane, no VGPR round trip,
  // tracked by ASYNCcnt. Flat LDS pointers carry the LDS offset in their low
  // 32 bits (ISA 10.2 aperture rule), so the int-cast forms a valid as(3)
  // address; as(1) addresses are numerically identical to flat global ones.
  {
    constexpr int CHUNKS = CN / 4;  // 150 x b128 per row, exact
    for (int idx = tid; idx < ROWS * CHUNKS; idx += 256) {
      int r = idx / CHUNKS, g = idx % CHUNKS;
      const float* src = logits + (size_t)(q0 + r) * CN + g * 4;
      float* dst = &tile[r][g * 4];
      __builtin_amdgcn_global_load_async_to_lds_b128(
          (__attribute__((address_space(1))) b128_t*)(uintptr_t)src,
          (__attribute__((address_space(3))) b128_t*)(uintptr_t)dst,
          /*offset=*/0, /*cpol=*/0);
    }
    if (tid < ROWS * (CPAD - CN)) {                 // zero the 8 pad columns
      int r = tid >> 3, c = CN + (tid & 7);
      tile[r][c] = 0.0f;
    }
#if __has_builtin(__builtin_amdgcn_s_wait_asynccnt)
    __builtin_amdgcn_s_wait_asynccnt(0);
#else
    asm volatile("s_wait_asynccnt 0" ::: "memory");
#endif
  }
#else
  for (int idx = tid; idx < ROWS * CPAD; idx += 256) {
    int r = idx / CPAD, c = idx % CPAD;
    tile[r][c] = (c < CN) ? logits[(q0 + r) * CN + c] : 0.0f;
  }
#endif
  __syncthreads();

  { // row max: 16 teams of 16 lanes
    int team = tid >> 4, lane = tid & 15;
    float m = -__builtin_inff();
    for (int c = lane; c < CN; c += 16) m = fmaxf(m, tile[team][c]);
    for (int s = 8; s > 0; s >>= 1) m = fmaxf(m, __shfl_xor(m, s, 16));
    if (lane == 0) rowmax[team] = m;
  }
  __syncthreads();

  for (int idx = tid; idx < ROWS * CPAD; idx += 256) {
    int r = idx / CPAD, c = idx % CPAD;
    tile[r][c] = (c < CN) ? __expf(tile[r][c] - rowmax[r]) : 0.0f;
  }
  __syncthreads();

  // Row sums on the matrix pipe: acc += A(16x4) * ones(4x16).
  // A layout (ISA 7.12.2): lanes 0-15 -> M=lane, K=0,1; lanes 16-31 -> M=lane-16, K=2,3.
  if ((tid >> 5) == 0) {
    const int lane = tid & 31;
    const int row  = lane & 15;
    const int kb   = (lane >> 4) * 2;
    v2f ones = {1.0f, 1.0f};
    v8f acc  = {0.f, 0.f, 0.f, 0.f, 0.f, 0.f, 0.f, 0.f};
    for (int ch = 0; ch < CPAD / 4; ++ch) {
      v2f a;
      a[0] = tile[row][ch * 4 + kb + 0];
      a[1] = tile[row][ch * 4 + kb + 1];
      acc = __builtin_amdgcn_wmma_f32_16x16x4_f32(
          /*neg_a=*/false, a, /*neg_b=*/false, ones,
          /*c_mod=*/(short)0, acc, /*reuse_a=*/false, /*reuse_b=*/false);
    }
    // C/D layout: VGPR r, lanes 0-15 -> M=r ; lanes 16-31 -> M=8+r. Sum replicated over N.
    if (lane == 0) {
      rowsum[0]=acc[0]; rowsum[1]=acc[1]; rowsum[2]=acc[2]; rowsum[3]=acc[3];
      rowsum[4]=acc[4]; rowsum[5]=acc[5]; rowsum[6]=acc[6]; rowsum[7]=acc[7];
    }
    if (lane == 16) {
      rowsum[8]=acc[0];  rowsum[9]=acc[1];  rowsum[10]=acc[2]; rowsum[11]=acc[3];
      rowsum[12]=acc[4]; rowsum[13]=acc[5]; rowsum[14]=acc[6]; rowsum[15]=acc[7];
    }
  }
  __syncthreads();

  for (int idx = tid; idx < ROWS * CN; idx += 256) {
    int r = idx / CN, c = idx % CN;
    float sc = tile[r][c] / rowsum[r] * sigm[r];
    if (sc > THR) {
      unsigned flat = (unsigned)(q0 + r) * CN + (unsigned)c;
      unsigned pos  = atomicAdd(count, 1u);
      if (pos < CAP)
        keys[pos] = ((uint64_t)__float_as_uint(sc) << 32) | (uint32_t)(~flat);
      if (flat < NFLAGS) above[flat] = 1;
    }
  }
}

// ------------------------------------------------------- bitonic sort ----
// Global pass for j >= 256 (pairs cross 256-element tiles).
__global__ void hoi_sort_pass(uint64_t* __restrict__ keys, int k, int j) {
  int i   = blockIdx.x * blockDim.x + threadIdx.x;
  int ixj = i ^ j;
  if (ixj > i) {
    uint64_t a = keys[i], b = keys[ixj];
    bool desc = ((i & k) == 0);
    if (desc ? (a < b) : (a > b)) { keys[i] = b; keys[ixj] = a; }
  }
}

// Fused final stages: all substages with j <= 128 run inside one 256-element
// LDS tile (128 threads, 2 elements/thread) -- 8 compare stages per global
// round trip instead of 8 separate kernel launches.
__global__ __launch_bounds__(128) void hoi_sort_fused(uint64_t* __restrict__ keys,
                                                      int k, int jstart) {
  __shared__ uint64_t s[256];
  const int base = blockIdx.x * 256;
  const int t = threadIdx.x;
  s[t]       = keys[base + t];
  s[t + 128] = keys[base + t + 128];
  __syncthreads();
  for (int j = jstart; j > 0; j >>= 1) {
    int i  = ((t & ~(j - 1)) << 1) | (t & (j - 1));  // bit j cleared
    int gi = base + i;
    uint64_t a = s[i], b = s[i + j];
    bool desc = ((gi & k) == 0);
    if (desc ? (a < b) : (a > b)) { s[i] = b; s[i + j] = a; }
    __syncthreads();
  }
  keys[base + t]       = s[t];
  keys[base + t + 128] = s[t + 128];
}

// ------------------------------------------------------- top-K extract ---
__global__ void hoi_extract(const uint64_t* __restrict__ keys, float* scoreTop,
                            int* qidx, int* cls, int* valid) {
  int kx = blockIdx.x * blockDim.x + threadIdx.x;
  if (kx >= KTOP) return;
  uint64_t key = keys[kx];
  int v = (key != 0ull);  // appended keys all have score > THR
  unsigned flat = ~(uint32_t)key;
  scoreTop[kx] = v ? __uint_as_float((uint32_t)(key >> 32)) : 0.0f;
  qidx[kx]     = v ? (int)(flat / (unsigned)CN) : 0;
  cls[kx]      = v ? (int)(flat % (unsigned)CN) : 0;
  valid[kx]    = v;
}

// Invalid top_k slots hold the smallest flat indices with score<=THR (-inf
// entries); rebuild their q_idx so pb.max()/ob.max() match the reference.
__global__ void hoi_fill_invalid(const unsigned* __restrict__ count,
                                 const unsigned char* __restrict__ above,
                                 int* __restrict__ qidx) {
  if (threadIdx.x != 0 || blockIdx.x != 0) return;
  int M = (int)*count; if (M > KTOP) M = KTOP;
  int need = KTOP - M, slot = M, f = 0;
  while (need > 0 && f < NFLAGS) {
    if (!above[f]) { qidx[slot++] = f / CN; --need; }
    ++f;
  }
}

// --------------------------------------------- gather boxes + coord max --
__global__ __launch_bounds__(256) void hoi_gather(
    const float* __restrict__ pred_boxes, const int* __restrict__ qidx,
    const int* __restrict__ ori_h, const int* __restrict__ ori_w,
    float* __restrict__ pb, float* __restrict__ ob,
    unsigned* __restrict__ maxPm, unsigned* __restrict__ maxOm) {
  __shared__ float smp[8], smo[8];
  int kx = blockIdx.x * 256 + threadIdx.x;
  float W = (float)(*ori_w), H = (float)(*ori_h);
  const float* b = pred_boxes + (size_t)qidx[kx] * 8;

  float px0 = fminf(fmaxf(b[0] - 0.5f * b[2], 0.f), 1.f) * W;
  float py0 = fminf(fmaxf(b[1] - 0.5f * b[3], 0.f), 1.f) * H;
  float px1 = fminf(fmaxf(b[0] + 0.5f * b[2], 0.f), 1.f) * W;
  float py1 = fminf(fmaxf(b[1] + 0.5f * b[3], 0.f), 1.f) * H;
  float ox0 = (b[4] - 0.5f * b[6]) * W;
  float oy0 = (b[5] - 0.5f * b[7]) * H;
  float ox1 = (b[4] + 0.5f * b[6]) * W;
  float oy1 = (b[5] + 0.5f * b[7]) * H;

  pb[kx*4+0]=px0; pb[kx*4+1]=py0; pb[kx*4+2]=px1; pb[kx*4+3]=py1;
  ob[kx*4+0]=ox0; ob[kx*4+1]=oy0; ob[kx*4+2]=ox1; ob[kx*4+3]=oy1;

  float mp = fmaxf(fmaxf(px0, py0), fmaxf(px1, py1));
  float mo = fmaxf(fmaxf(ox0, oy0), fmaxf(ox1, oy1));
  for (int s = 16; s > 0; s >>= 1) {
    mp = fmaxf(mp, __shfl_xor(mp, s, 32));
    mo = fmaxf(mo, __shfl_xor(mo, s, 32));
  }
  if ((threadIdx.x & 31) == 0) { smp[threadIdx.x >> 5] = mp; smo[threadIdx.x >> 5] = mo; }
  __syncthreads();
  if (threadIdx.x == 0) {
    float ap = smp[0], ao = smo[0];
    for (int i = 1; i < 8; ++i) { ap = fmaxf(ap, smp[i]); ao = fmaxf(ao, smo[i]); }
    atomicMax(maxPm, fmap(ap));
    atomicMax(maxOm, fmap(ao));
  }
}

// -------------------------------------------- class-offset boxes (NMS) ---
__global__ void hoi_offset(const float* __restrict__ pb, const float* __restrict__ ob,
                           const int* __restrict__ cls,
                           const unsigned* __restrict__ maxPm,
                           const unsigned* __restrict__ maxOm,
                           float* __restrict__ pbo, float* __restrict__ obo) {
  int kx = blockIdx.x * blockDim.x + threadIdx.x;
  if (kx >= KTOP) return;
  float c   = (float)cls[kx];
  float oP  = c * (funmap(*maxPm) + 1.0f);
  float oO  = c * (funmap(*maxOm) + 1.0f);
  for (int d = 0; d < 4; ++d) {
    pbo[kx*4+d] = pb[kx*4+d] + oP;
    obo[kx*4+d] = ob[kx*4+d] + oO;
  }
}

// ---------------------------------------------------- NMS bitmask --------
__global__ __launch_bounds__(NCHUNK) void hoi_nms_mask(
    const float* __restrict__ boxes, const int* __restrict__ valid,
    unsigned* __restrict__ mask) {
  const int i = blockIdx.x, t = threadIdx.x;
  float ix0 = boxes[i*4+0], iy0 = boxes[i*4+1], ix1 = boxes[i*4+2], iy1 = boxes[i*4+3];
  float ia  = (ix1 - ix0) * (iy1 - iy0);
  unsigned bits = 0;
  int jbase = t * 32;
  for (int jj = 0; jj < 32; ++jj) {
    int j = jbase + jj;
    if (j > i && valid[j]) {
      float jx0 = boxes[j*4+0], jy0 = boxes[j*4+1], jx1 = boxes[j*4+2], jy1 = boxes[j*4+3];
      float ja = (jx1 - jx0) * (jy1 - jy0);
      float w  = fmaxf(fminf(ix1, jx1) - fmaxf(ix0, jx0), 0.f);
      float h  = fmaxf(fminf(iy1, jy1) - fmaxf(iy0, jy0), 0.f);
      float inter = w * h;
      float iou = inter / fmaxf(ia + ja - inter, 1e-9f);
      if (iou > IOU_THR) bits |= 1u << jj;
    }
  }
  mask[(size_t)i * NCHUNK + t] = bits;
}

// serial greedy scan (matches fori_loop suppression order exactly)
__global__ __launch_bounds__(NCHUNK) void hoi_nms_scan(
    const unsigned* __restrict__ mask, const int* __restrict__ valid,
    int* __restrict__ keep) {
  __shared__ unsigned removed[NCHUNK];
  const int t = threadIdx.x;
  removed[t] = 0u;
  __syncthreads();
  for (int i = 0; i < KTOP; ++i) {
    bool kp = valid[i] && !((removed[i >> 5] >> (i & 31)) & 1u);
    __syncthreads();
    if (kp) removed[t] |= mask[(size_t)i * NCHUNK + t];
    __syncthreads();
  }
  for (int i = t; i < KTOP; i += NCHUNK)
    keep[i] = valid[i] && !((removed[i >> 5] >> (i & 31)) & 1u);
}

// --------------------------------------------------------- outputs ------
__global__ void hoi_finalize(const float* __restrict__ scoreTop,
                             const int* __restrict__ cls,
                             const int* __restrict__ keepP,
                             const int* __restrict__ keepO,
                             const float* __restrict__ pb,
                             const float* __restrict__ ob,
                             const int* __restrict__ mapper,
                             float* __restrict__ out) {
  int kx = blockIdx.x * blockDim.x + threadIdx.x;
  if (kx >= KTOP) return;
  int f = keepP[kx] | keepO[kx];                 // keep arrays already include valid
  float m = f ? 1.0f : 0.0f;
  out[kx]              = f ? (float)mapper[cls[kx]] : -1.0f;   // hoi_ids
  out[KTOP + kx]       = f ? scoreTop[kx] : 0.0f;              // scores
  for (int d = 0; d < 4; ++d) {
    out[2*KTOP + kx*8 + d]     = pb[kx*4+d] * m;               // boxes[:, :4]
    out[2*KTOP + kx*8 + 4 + d] = ob[kx*4+d] * m;               // boxes[:, 4:]
  }
  out[10*KTOP + kx] = m;                                       // filt
}

// ---------------------------------------------------------------------------
extern "C" void kernel_launch(void* const* d_in, const int* in_sizes, int n_in,
                              void* d_out, int out_size, void* d_ws, size_t ws_size,
                              hipStream_t stream) {
  (void)in_sizes; (void)n_in; (void)out_size; (void)ws_size;
  const float* pred_boxes = (const float*)d_in[0];
  const float* pred_logits= (const float*)d_in[1];
  const float* box_scores = (const float*)d_in[2];
  const int*   hoi_mapper = (const int*)d_in[3];
  const int*   ori_h      = (const int*)d_in[4];
  const int*   ori_w      = (const int*)d_in[5];
  float* out = (float*)d_out;

  // workspace layout (bytes); total ~8.8 MB
  char* ws = (char*)d_ws;
  size_t off = 0;
  uint64_t* keys   = (uint64_t*)(ws + off); off += (size_t)CAP * 8;
  unsigned* count  = (unsigned*)(ws + off); off += 16;
  unsigned* maxPm  = (unsigned*)(ws + off); off += 4;
  unsigned* maxOm  = (unsigned*)(ws + off); off += 8;   // keep 4B align slack
  unsigned char* above = (unsigned char*)(ws + off); off += NFLAGS;
  float* scoreTop  = (float*)(ws + off); off += KTOP * 4;
  int*   qidx      = (int*)  (ws + off); off += KTOP * 4;
  int*   cls       = (int*)  (ws + off); off += KTOP * 4;
  int*   valid     = (int*)  (ws + off); off += KTOP * 4;
  float* pb        = (float*)(ws + off); off += KTOP * 16;
  float* ob        = (float*)(ws + off); off += KTOP * 16;
  float* pbo       = (float*)(ws + off); off += KTOP * 16;
  float* obo       = (float*)(ws + off); off += KTOP * 16;
  unsigned* maskP  = (unsigned*)(ws + off); off += (size_t)KTOP * NCHUNK * 4;
  unsigned* maskO  = (unsigned*)(ws + off); off += (size_t)KTOP * NCHUNK * 4;
  int*   keepP     = (int*)(ws + off); off += KTOP * 4;
  int*   keepO     = (int*)(ws + off); off += KTOP * 4;

  hipLaunchKernelGGL(hoi_init, dim3(CAP / 256), dim3(256), 0, stream,
                     keys, count, maxPm, maxOm, (unsigned*)above);

  hipLaunchKernelGGL(hoi_score, dim3(QN / ROWS), dim3(256), 0, stream,
                     pred_logits, box_scores, keys, count, above);

  // bitonic sort, descending: global passes for j>=256, fused LDS kernel for j<=128
  for (int k = 2; k <= (int)CAP; k <<= 1) {
    int j = k >> 1;
    for (; j >= 256; j >>= 1)
      hipLaunchKernelGGL(hoi_sort_pass, dim3(CAP / 256), dim3(256), 0, stream,
                         keys, k, j);
    hipLaunchKernelGGL(hoi_sort_fused, dim3(CAP / 256), dim3(128), 0, stream,
                       keys, k, j);
  }

  hipLaunchKernelGGL(hoi_extract, dim3(KTOP / 256), dim3(256), 0, stream,
                     keys, scoreTop, qidx, cls, valid);
  hipLaunchKernelGGL(hoi_fill_invalid, dim3(1), dim3(32), 0, stream,
                     count, above, qidx);
  hipLaunchKernelGGL(hoi_gather, dim3(KTOP / 256), dim3(256), 0, stream,
                     pred_boxes, qidx, ori_h, ori_w, pb, ob, maxPm, maxOm);
  hipLaunchKernelGGL(hoi_offset, dim3(KTOP / 256), dim3(256), 0, stream,
                     pb, ob, cls, maxPm, maxOm, pbo, obo);

  hipLaunchKernelGGL(hoi_nms_mask, dim3(KTOP), dim3(NCHUNK), 0, stream, pbo, valid, maskP);
  hipLaunchKernelGGL(hoi_nms_mask, dim3(KTOP), dim3(NCHUNK), 0, stream, obo, valid, maskO);
  hipLaunchKernelGGL(hoi_nms_scan, dim3(1), dim3(NCHUNK), 0, stream, maskP, valid, keepP);
  hipLaunchKernelGGL(hoi_nms_scan, dim3(1), dim3(NCHUNK), 0, stream, maskO, valid, keepO);

  hipLaunchKernelGGL(hoi_finalize, dim3(KTOP / 256), dim3(256), 0, stream,
                     scoreTop, cls, keepP, keepO, pb, ob, hoi_mapper, out);
}